// HyperMTANPro_15281493639482
// MI455X (gfx1250) — compile-verified
//
#include <hip/hip_runtime.h>
#include <math.h>

// ---------------------------------------------------------------------------
// HyperMTAN fused forward for MI455X (gfx1250, wave32, WMMA).
//
// Per-sample hyper-generated matmuls  x_b @ reshape(hyper_b@Wh+bh)  are
// rewritten as dense GEMMs  F @ [Wh;bh]  with F_b[h*n+j] = x_b[j] * ha_b[h],
// ha = [hyper_x, 1]; generated biases fold in as 11 extra K rows [Wb;bb].
// All GEMMs use V_WMMA_F32_16X16X4_F32 (full fp32 like the reference).
//
// Workgroup = 8 wave32 = 32-sample tile (TWO 16-row WMMA M-tiles per wave):
// each streamed B-fragment (weights, L2-resident) feeds two v_wmma issues.
// K loop is tiled h-outer / n-inner: ha factors hoist out of the inner loop,
// the bh block (h==10, ha==1) needs no multiplies, and each segment streams
// from a single base pointer (no per-iteration Wh/bh select).
// ---------------------------------------------------------------------------

typedef __attribute__((ext_vector_type(2))) float v2f;
typedef __attribute__((ext_vector_type(8))) float v8f;

#define XSTR   260   // LDS row stride in floats (260 % 64 == 4 -> conflict-free column reads)
#define NWAVES 8
#define SB     32    // samples per workgroup (two WMMA M-tiles)

__device__ __forceinline__ v8f wmma4(v2f a, v2f b, v8f c) {
  // D = A(16x4 f32) * B(4x16 f32) + C(16x16 f32)
  return __builtin_amdgcn_wmma_f32_16x16x4_f32(
      /*neg_a=*/false, a, /*neg_b=*/false, b,
      /*c_mod=*/(short)0, c, /*reuse_a=*/false, /*reuse_b=*/false);
}

// One hypernetwork layer: out[SB, m] = act( F[SB, 11n] @ [Wh;bh] + ha @ [Wb;bb] )
// NSHIFT = log2(n_dim). MODE: 0 = relu -> out_lds
//                             1 = sigmoid(.) * mulg -> out_lds   (mask * g1 fused)
//                             2 = raw -> out_glob (row-major, width MDIM)
template <int NSHIFT, int MODE, int MDIM>
__device__ __forceinline__ void hyper_layer(
    const float* __restrict__ x_lds, const float* __restrict__ ha_lds,
    const float* __restrict__ Wh, const float* __restrict__ bh,
    const float* __restrict__ Wb, const float* __restrict__ bb,
    float* __restrict__ out_lds, const float* __restrict__ mulg,
    float* __restrict__ out_glob, int wave, int half, int l) {
  const int n_dim = 1 << NSHIFT;
  const int ntiles = (MDIM + 15) >> 4;
  const int r0 = l;            // A row for M-tile 0 (samples 0..15)
  const int r1 = l + 16;       // A row for M-tile 1 (samples 16..31)

  for (int tile = wave; tile < ntiles; tile += NWAVES) {  // wave-uniform -> EXEC full
    const int ncol = tile << 4;
    const int col  = ncol + l;
    const int colc = (col < MDIM) ? col : (MDIM - 1);     // clamp for m=8 tail tile

    v8f c0 = {};
    v8f c1 = {};

    // ---- main GEMM: h-outer (ha factor hoisted), n-inner over Wh rows ----
    for (int h = 0; h < 10; ++h) {
      const float hx0 = ha_lds[r0 * 12 + h];
      const float hx1 = ha_lds[r1 * 12 + h];
      const float* p = Wh + (h * n_dim + 2 * half) * MDIM + colc;
      for (int n0 = 0; n0 < n_dim; n0 += 4) {
        const int na = n0 + 2 * half;
        v2f a0, a1, b;
        a0.x = x_lds[r0 * XSTR + na]     * hx0;
        a0.y = x_lds[r0 * XSTR + na + 1] * hx0;
        a1.x = x_lds[r1 * XSTR + na]     * hx1;
        a1.y = x_lds[r1 * XSTR + na + 1] * hx1;
        b.x = p[0];
        b.y = p[MDIM];
        p += 4 * MDIM;
        c0 = wmma4(a0, b, c0);           // shared B feeds both M-tiles
        c1 = wmma4(a1, b, c1);
      }
    }

    // ---- bh block (h == 10, ha == 1): F = x directly, no multiplies ----
    {
      const float* p = bh + (2 * half) * MDIM + colc;
      for (int n0 = 0; n0 < n_dim; n0 += 4) {
        const int na = n0 + 2 * half;
        v2f a0, a1, b;
        a0.x = x_lds[r0 * XSTR + na];
        a0.y = x_lds[r0 * XSTR + na + 1];
        a1.x = x_lds[r1 * XSTR + na];
        a1.y = x_lds[r1 * XSTR + na + 1];
        b.x = p[0];
        b.y = p[MDIM];
        p += 4 * MDIM;
        c0 = wmma4(a0, b, c0);
        c1 = wmma4(a1, b, c1);
      }
    }

    // ---- generated-bias GEMM: 12 K rows of [Wb ; bb ; 0] against ha ----
#pragma unroll
    for (int k0 = 0; k0 < 12; k0 += 4) {
      const int ka = k0 + 2 * half;
      v2f a0, a1, b;
      a0.x = ha_lds[r0 * 12 + ka];       // ha[...10]=1, ha[...11]=0
      a0.y = ha_lds[r0 * 12 + ka + 1];
      a1.x = ha_lds[r1 * 12 + ka];
      a1.y = ha_lds[r1 * 12 + ka + 1];
      b.x = (ka < 10) ? Wb[ka * MDIM + colc] : ((ka == 10) ? bb[colc] : 0.0f);
      const int kb = ka + 1;
      b.y = (kb < 10) ? Wb[kb * MDIM + colc] : ((kb == 10) ? bb[colc] : 0.0f);
      c0 = wmma4(a0, b, c0);
      c1 = wmma4(a1, b, c1);
    }

    // ---- epilogue (both M-tiles) ----
#pragma unroll
    for (int r = 0; r < 8; ++r) {
      const int row0 = r + 8 * half;
      const int row1 = row0 + 16;
      float v0 = c0[r];
      float v1 = c1[r];
      if (MODE == 1) {
        v0 = 1.0f / (1.0f + __expf(-v0));
        v1 = 1.0f / (1.0f + __expf(-v1));
        v0 *= mulg[row0 * XSTR + col];   // fuse mask * g1
        v1 *= mulg[row1 * XSTR + col];
        out_lds[row0 * XSTR + col] = v0;
        out_lds[row1 * XSTR + col] = v1;
      } else if (MODE == 0) {
        out_lds[row0 * XSTR + col] = fmaxf(v0, 0.0f);
        out_lds[row1 * XSTR + col] = fmaxf(v1, 0.0f);
      } else {
        if (col < MDIM) {
          out_glob[row0 * MDIM + col] = v0;
          out_glob[row1 * MDIM + col] = v1;
        }
      }
    }
  }
}

__global__ __launch_bounds__(256) void hypermtan_fused_kernel(
    const float* __restrict__ hyper_x, const float* __restrict__ mlp_x,
    const float* __restrict__ G0_w, const float* __restrict__ G0_b,
    const float* __restrict__ G1_w, const float* __restrict__ G1_b,
    const float* __restrict__ Wmw0, const float* __restrict__ bmw0,
    const float* __restrict__ Wmb0, const float* __restrict__ bmb0,
    const float* __restrict__ Wmw1, const float* __restrict__ bmw1,
    const float* __restrict__ Wmb1, const float* __restrict__ bmb1,
    const float* __restrict__ Wmw2, const float* __restrict__ bmw2,
    const float* __restrict__ Wmb2, const float* __restrict__ bmb2,
    const float* __restrict__ Wmw3, const float* __restrict__ bmw3,
    const float* __restrict__ Wmb3, const float* __restrict__ bmb3,
    const float* __restrict__ Wew,  const float* __restrict__ bew,
    const float* __restrict__ Web,  const float* __restrict__ beb,
    const float* __restrict__ Wlw,  const float* __restrict__ blw,
    const float* __restrict__ Wlb,  const float* __restrict__ blb,
    float* __restrict__ out) {
  __shared__ float ha[SB * 12];      // [hyper_x, 1, 0] per sample
  __shared__ float x0[SB * 12];      // mlp_x zero-padded to 12
  __shared__ float g0[SB * XSTR];
  __shared__ float g1[SB * XSTR];
  __shared__ float bufA[SB * XSTR];
  __shared__ float bufB[SB * XSTR];

  const int tid  = threadIdx.x;
  const int wave = tid >> 5;
  const int lane = tid & 31;
  const int half = lane >> 4;
  const int l    = lane & 15;
  const int b0   = blockIdx.x * SB;  // first sample of this tile

  // Stage in hyper/mlp inputs (augmented).
  for (int i = tid; i < SB * 12; i += 256) {
    const int r = i / 12, cidx = i - r * 12;
    ha[i] = (cidx < 10) ? hyper_x[(b0 + r) * 10 + cidx] : ((cidx == 10) ? 1.0f : 0.0f);
    x0[i] = (cidx < 9) ? mlp_x[(b0 + r) * 9 + cidx] : 0.0f;
  }
  __syncthreads();

  // ---- g0 = relu(mlp_x @ G0_w + G0_b)  [SB,9]x[9,256], K padded to 12 ----
  for (int tile = wave; tile < 16; tile += NWAVES) {
    const int ncol = tile << 4;
    v8f c0 = {};
    v8f c1 = {};
#pragma unroll
    for (int k0 = 0; k0 < 12; k0 += 4) {
      const int ka = k0 + 2 * half;
      v2f a0, a1, b;
      a0.x = x0[l * 12 + ka];
      a0.y = x0[l * 12 + ka + 1];
      a1.x = x0[(l + 16) * 12 + ka];
      a1.y = x0[(l + 16) * 12 + ka + 1];
      b.x = (ka < 9)     ? G0_w[ka * 256 + ncol + l]       : 0.0f;
      b.y = (ka + 1 < 9) ? G0_w[(ka + 1) * 256 + ncol + l] : 0.0f;
      c0 = wmma4(a0, b, c0);
      c1 = wmma4(a1, b, c1);
    }
    const float bias = G0_b[ncol + l];
#pragma unroll
    for (int r = 0; r < 8; ++r) {
      g0[(r + 8 * half) * XSTR + ncol + l]      = fmaxf(c0[r] + bias, 0.0f);
      g0[(r + 8 * half + 16) * XSTR + ncol + l] = fmaxf(c1[r] + bias, 0.0f);
    }
  }
  __syncthreads();

  // ---- g1 = relu(g0 @ G1_w + G1_b)  [SB,256]x[256,256] ----
  for (int tile = wave; tile < 16; tile += NWAVES) {
    const int ncol = tile << 4;
    v8f c0 = {};
    v8f c1 = {};
    const float* p = G1_w + (2 * half) * 256 + ncol + l;
    for (int k0 = 0; k0 < 256; k0 += 4) {
      const int ka = k0 + 2 * half;
      v2f a0, a1, b;
      a0.x = g0[l * XSTR + ka];
      a0.y = g0[l * XSTR + ka + 1];
      a1.x = g0[(l + 16) * XSTR + ka];
      a1.y = g0[(l + 16) * XSTR + ka + 1];
      b.x = p[0];
      b.y = p[256];
      p += 4 * 256;
      c0 = wmma4(a0, b, c0);
      c1 = wmma4(a1, b, c1);
    }
    const float bias = G1_b[ncol + l];
#pragma unroll
    for (int r = 0; r < 8; ++r) {
      g1[(r + 8 * half) * XSTR + ncol + l]      = fmaxf(c0[r] + bias, 0.0f);
      g1[(r + 8 * half + 16) * XSTR + ncol + l] = fmaxf(c1[r] + bias, 0.0f);
    }
  }
  __syncthreads();

  // ---- hypernetwork chain (each one dense GEMM via outer-product features) ----
  hyper_layer<8, 0, 128>(g0,   ha, Wmw0, bmw0, Wmb0, bmb0, bufA, nullptr, nullptr, wave, half, l);
  __syncthreads();
  hyper_layer<7, 0, 128>(bufA, ha, Wmw1, bmw1, Wmb1, bmb1, bufB, nullptr, nullptr, wave, half, l);
  __syncthreads();
  hyper_layer<7, 0, 128>(bufB, ha, Wmw2, bmw2, Wmb2, bmb2, bufA, nullptr, nullptr, wave, half, l);
  __syncthreads();
  // mask = sigmoid(...), fused with (mask * g1)
  hyper_layer<7, 1, 256>(bufA, ha, Wmw3, bmw3, Wmb3, bmb3, bufB, g1, nullptr, wave, half, l);
  __syncthreads();
  // extractor
  hyper_layer<8, 0, 64>(bufB, ha, Wew, bew, Web, beb, bufA, nullptr, nullptr, wave, half, l);
  __syncthreads();
  // last layer -> global, m=8
  hyper_layer<6, 2, 8>(bufA, ha, Wlw, blw, Wlb, blb, nullptr, nullptr, out + (size_t)b0 * 8, wave, half, l);
}

extern "C" void kernel_launch(void* const* d_in, const int* in_sizes, int n_in,
                              void* d_out, int out_size, void* d_ws, size_t ws_size,
                              hipStream_t stream) {
  (void)in_sizes; (void)n_in; (void)d_ws; (void)ws_size; (void)out_size;
  const float* hyper_x = (const float*)d_in[0];
  const float* mlp_x   = (const float*)d_in[1];
  const float* G0_w = (const float*)d_in[2];  const float* G0_b = (const float*)d_in[3];
  const float* G1_w = (const float*)d_in[4];  const float* G1_b = (const float*)d_in[5];
  const float* Wmw0 = (const float*)d_in[6];  const float* bmw0 = (const float*)d_in[7];
  const float* Wmb0 = (const float*)d_in[8];  const float* bmb0 = (const float*)d_in[9];
  const float* Wmw1 = (const float*)d_in[10]; const float* bmw1 = (const float*)d_in[11];
  const float* Wmb1 = (const float*)d_in[12]; const float* bmb1 = (const float*)d_in[13];
  const float* Wmw2 = (const float*)d_in[14]; const float* bmw2 = (const float*)d_in[15];
  const float* Wmb2 = (const float*)d_in[16]; const float* bmb2 = (const float*)d_in[17];
  const float* Wmw3 = (const float*)d_in[18]; const float* bmw3 = (const float*)d_in[19];
  const float* Wmb3 = (const float*)d_in[20]; const float* bmb3 = (const float*)d_in[21];
  const float* Wew  = (const float*)d_in[22]; const float* bew  = (const float*)d_in[23];
  const float* Web  = (const float*)d_in[24]; const float* beb  = (const float*)d_in[25];
  const float* Wlw  = (const float*)d_in[26]; const float* blw  = (const float*)d_in[27];
  const float* Wlb  = (const float*)d_in[28]; const float* blb  = (const float*)d_in[29];
  float* out = (float*)d_out;

  hipLaunchKernelGGL(hypermtan_fused_kernel, dim3(1024 / SB), dim3(256), 0, stream,
                     hyper_x, mlp_x, G0_w, G0_b, G1_w, G1_b,
                     Wmw0, bmw0, Wmb0, bmb0, Wmw1, bmw1, Wmb1, bmb1,
                     Wmw2, bmw2, Wmb2, bmb2, Wmw3, bmw3, Wmb3, bmb3,
                     Wew, bew, Web, beb, Wlw, blw, Wlb, blb, out);
}